// Izhikevich_SNN_65206193487989
// MI455X (gfx1250) — compile-verified
//
#include <hip/hip_runtime.h>

// Izhikevich SNN, T=8192 steps, N=1024 neurons.
// Phase A: I[t][n] = sum_k X[t][k] * W1[n][k]  -> fp32 WMMA GEMM (v_wmma_f32_16x16x4_f32)
//          with double-buffered LDS staging via GLOBAL_LOAD_ASYNC_TO_LDS_B128 (ASYNCcnt)
//          when the toolchain exposes the async builtins; coalesced sync fallback else.
//          I is staged into the states[t][0][n] slots of d_out (overwritten by phase B).
// Phase B: per-neuron sequential scan over t (32 blocks x 32 lanes = 1024 neurons),
//          wave-shuffle reduction of W2[n]*s[n] -> per-block partials in d_ws.
// Phase C: q[t] = sum of 32 block partials (parallel), then serial dm recurrence.

#define T_STEPS 8192
#define N_NEUR  1024
#define KC      16           // K-chunk staged in LDS per double-buffer step
#define LDS_PITCH 20         // floats per LDS row (16 data + 4 pad): 20*r mod 64 is
                             // injective for r in [0,16) -> conflict-free ds_load_b64

typedef __attribute__((ext_vector_type(2))) float v2f;
typedef __attribute__((ext_vector_type(4))) float v4f;
typedef __attribute__((ext_vector_type(8))) float v8f;
typedef __attribute__((ext_vector_type(4))) int   v4i;

#define AS1 __attribute__((address_space(1)))
#define AS3 __attribute__((address_space(3)))

#if defined(__has_builtin)
#if __has_builtin(__builtin_amdgcn_global_load_async_to_lds_b128) && \
    __has_builtin(__builtin_amdgcn_s_wait_asynccnt)
#define SNN_ASYNC_COPY 1
#endif
#endif
#ifndef SNN_ASYNC_COPY
#define SNN_ASYNC_COPY 0
#endif

#if SNN_ASYNC_COPY
#define ASYNC_WAIT() __builtin_amdgcn_s_wait_asynccnt(0)
#else
#define ASYNC_WAIT() ((void)0)
#endif

// Builtin signature (from hipcc diagnostic): param0 is a global-AS (printed as
// "__device__") non-const v4i*, so cast through integers to the exact types.
__device__ __forceinline__ AS3 v4i* lds_v4i(float* p) {
    // Generic LDS pointer: addr[31:0] is the LDS offset (ISA 10.2 aperture rules).
    return (AS3 v4i*)(unsigned)(unsigned long long)p;
}
__device__ __forceinline__ AS1 v4i* gbl_v4i(const float* p) {
    return (AS1 v4i*)(unsigned long long)p;
}

// Stage one 128x16 A-tile and 128x16 B-tile into LDS.
// 512 b128 chunks per tile; thread th moves chunks {i*128+th}: 4 consecutive lanes
// cover one 64B row segment -> fully-consumed cache lines.
__device__ __forceinline__ void stage_tile(const float* gA, const float* gB,
                                           float* sA, float* sB, int th)
{
#pragma unroll
    for (int i = 0; i < 4; ++i) {
        const int cid = i * 128 + th;   // 0..511
        const int row = cid >> 2;       // 0..127
        const int c   = cid & 3;        // 16B chunk within the 16-float row
#if SNN_ASYNC_COPY
        __builtin_amdgcn_global_load_async_to_lds_b128(
            gbl_v4i(gA + (size_t)row * N_NEUR + c * 4),
            lds_v4i(sA + row * LDS_PITCH + c * 4), 0, 0);
        __builtin_amdgcn_global_load_async_to_lds_b128(
            gbl_v4i(gB + (size_t)row * N_NEUR + c * 4),
            lds_v4i(sB + row * LDS_PITCH + c * 4), 0, 0);
#else
        *(v4f*)(sA + row * LDS_PITCH + c * 4) =
            *(const v4f*)(gA + (size_t)row * N_NEUR + c * 4);
        *(v4f*)(sB + row * LDS_PITCH + c * 4) =
            *(const v4f*)(gB + (size_t)row * N_NEUR + c * 4);
#endif
    }
}

// ---------------------------------------------------------------------------
// Phase A: GEMM  I = X @ W1^T  using V_WMMA_F32_16X16X4_F32.
// Block = 128 threads = 4 waves arranged 2x2; block tile 128(M) x 128(N).
// Wave tile 64x64 = 4x4 subtiles of 16x16 (128 accumulator VGPRs).
//
// f32 WMMA fragment layouts (ISA 7.12.2):
//   A (16x4, MxK): lanes 0-15 hold {K=k0, k0+1}, lanes 16-31 hold {K=k0+2, k0+3},
//                  row M = lane&15  -> per-lane contiguous float2 (ds_load_b64).
//   B (4x16, KxN): rows striped across lanes within a VGPR (same half split);
//                  B[k][n] = W1[n][k] -> per-lane contiguous float2 of a W1 row.
//   C/D (16x16):   VGPR v: lanes 0-15 -> M=v, lanes 16-31 -> M=v+8; N = lane&15.
// ---------------------------------------------------------------------------
__global__ __launch_bounds__(128) void snn_gemm_wmma(
    const float* __restrict__ X,        // [T, N]
    const float* __restrict__ W1,       // [N, N] row-major
    float* __restrict__ statesOut)      // [T, 2, N]; I -> states[t][0][n]
{
    __shared__ float sA[2][128 * LDS_PITCH];   // 10 KB each
    __shared__ float sB[2][128 * LDS_PITCH];   // total 40 KB of 320 KB

    const int th   = threadIdx.x;
    const int lane = th & 31;
    const int wave = th >> 5;                   // 0..3
    const int wr   = wave >> 1, wc = wave & 1;  // 2x2 wave grid
    const int r    = lane & 15;
    const int half = lane >> 4;

    const int m_blk = blockIdx.y * 128;
    const int n_blk = blockIdx.x * 128;
    const int m0 = wr * 64;                     // wave offset inside block tile
    const int n0 = wc * 64;

    const float* gA = X  + (size_t)m_blk * N_NEUR;
    const float* gB = W1 + (size_t)n_blk * N_NEUR;

    v8f acc[4][4] = {};

    // Prologue: stage chunk 0
    stage_tile(gA, gB, sA[0], sB[0], th);
    ASYNC_WAIT();
    __syncthreads();

    for (int kc = 0; kc < N_NEUR / KC; ++kc) {
        const int cur = kc & 1;
        // Prefetch next chunk into the other buffer (async engine overlaps WMMA).
        if (kc + 1 < N_NEUR / KC)
            stage_tile(gA + (kc + 1) * KC, gB + (kc + 1) * KC,
                       sA[cur ^ 1], sB[cur ^ 1], th);

        const float* A = sA[cur];
        const float* B = sB[cur];
#pragma unroll
        for (int kk = 0; kk < KC; kk += 4) {
            v2f af[4], bf[4];
#pragma unroll
            for (int i = 0; i < 4; ++i)
                af[i] = *(const v2f*)(A + (m0 + i * 16 + r) * LDS_PITCH + kk + 2 * half);
#pragma unroll
            for (int j = 0; j < 4; ++j)
                bf[j] = *(const v2f*)(B + (n0 + j * 16 + r) * LDS_PITCH + kk + 2 * half);
#pragma unroll
            for (int i = 0; i < 4; ++i)
#pragma unroll
                for (int j = 0; j < 4; ++j)
                    acc[i][j] = __builtin_amdgcn_wmma_f32_16x16x4_f32(
                        /*neg_a=*/false, af[i], /*neg_b=*/false, bf[j],
                        /*c_mod=*/(short)0, acc[i][j],
                        /*reuse_a=*/false, /*reuse_b=*/false);
        }
        ASYNC_WAIT();       // next chunk's async copies complete
        __syncthreads();    // all waves staged + done reading current buffer
    }

    // Store I into the v-slot of states: states[row][0][col]
#pragma unroll
    for (int i = 0; i < 4; ++i)
#pragma unroll
        for (int j = 0; j < 4; ++j)
#pragma unroll
            for (int v = 0; v < 8; ++v) {
                const int row = m_blk + m0 + i * 16 + v + 8 * half;
                const int col = n_blk + n0 + j * 16 + r;
                statesOut[(size_t)row * (2 * N_NEUR) + col] = acc[i][j][v];
            }
}

// ---------------------------------------------------------------------------
// Phase B: sequential Izhikevich scan. 32 blocks x 32 threads (1 wave/block),
// thread = one neuron for all 8192 steps. Reads I from states[t][0][n] and
// overwrites with v_out; writes u_out and spikes; wave-reduces W2[n]*s.
// ---------------------------------------------------------------------------
__global__ __launch_bounds__(32) void snn_scan(
    const float* __restrict__ state,   // [2, N]
    const float* __restrict__ a_,
    const float* __restrict__ b_,
    const float* __restrict__ c_,
    const float* __restrict__ d_,
    const float* __restrict__ thresh,
    const float* __restrict__ dt_,
    const float* __restrict__ W2,      // [1, N]
    float* __restrict__ outputs,       // [T, N]
    float* __restrict__ states,        // [T, 2, N] (I pre-staged in v-slot)
    float* __restrict__ p_part)        // [T, 32] per-block partial W2@s
{
    const int lane = threadIdx.x;                 // 0..31
    const int n    = blockIdx.x * 32 + lane;      // neuron id

    float v = state[n];
    float u = state[N_NEUR + n];
    const float an = a_[n], bn = b_[n], cn = c_[n], dn = d_[n];
    const float th = thresh[n], dtn = dt_[n], w2n = W2[n];

    for (int t = 0; t < T_STEPS; ++t) {
        const size_t sbase = (size_t)t * (2 * N_NEUR);
        const float I = states[sbase + n];

        const float vn = v + dtn * (0.04f * v * v + 5.0f * v + 140.0f - u + I);
        const float un = u + dtn * an * (bn * v - u);
        const float s  = ((vn - th) > 0.0f) ? 1.0f : 0.0f;
        v = vn * (1.0f - s) + cn * s;
        u = un + dn * s;

        outputs[(size_t)t * N_NEUR + n] = s;
        states[sbase + n]              = v;
        states[sbase + N_NEUR + n]     = u;

        // wave32 tree reduction of w2n * s
        float part = w2n * s;
#pragma unroll
        for (int off = 16; off > 0; off >>= 1)
            part += __shfl_down(part, off, 32);
        if (lane == 0)
            p_part[t * 32 + blockIdx.x] = part;
    }
}

// ---------------------------------------------------------------------------
// Phase C1: q[t] = sum over 32 block partials (fully parallel, deterministic).
// ---------------------------------------------------------------------------
__global__ __launch_bounds__(256) void snn_reduce_q(
    const float* __restrict__ p_part, float* __restrict__ q)
{
    const int t = blockIdx.x * 256 + threadIdx.x;
    if (t < T_STEPS) {
        float s = 0.0f;
#pragma unroll
        for (int b = 0; b < 32; ++b)
            s += p_part[t * 32 + b];
        q[t] = s;
    }
}

// ---------------------------------------------------------------------------
// Phase C2: serial leaky-integrator decode: dm_t = leak*dm_{t-1} + q_t.
// One dependent FMA per step; loads of q[] are independent and prefetchable.
// ---------------------------------------------------------------------------
__global__ void snn_dm_scan(const float* __restrict__ q,
                            const float* __restrict__ leak,
                            float* __restrict__ decoded)
{
    if (threadIdx.x == 0 && blockIdx.x == 0) {
        const float lk = leak[0];
        float dm = 0.0f;
        for (int t = 0; t < T_STEPS; ++t) {
            dm = fmaf(lk, dm, q[t]);
            decoded[t] = dm;
        }
    }
}

// ---------------------------------------------------------------------------
extern "C" void kernel_launch(void* const* d_in, const int* in_sizes, int n_in,
                              void* d_out, int out_size, void* d_ws, size_t ws_size,
                              hipStream_t stream) {
    // setup_inputs() order:
    const float* X      = (const float*)d_in[0];   // [8192,1024]
    const float* state  = (const float*)d_in[1];   // [2,1024]
    const float* W1     = (const float*)d_in[2];   // [1024,1024]
    const float* W2     = (const float*)d_in[3];   // [1,1024]
    const float* a_     = (const float*)d_in[4];
    const float* b_     = (const float*)d_in[5];
    const float* c_     = (const float*)d_in[6];
    const float* d_     = (const float*)d_in[7];
    const float* thresh = (const float*)d_in[8];
    const float* dt_    = (const float*)d_in[9];
    const float* leak   = (const float*)d_in[10];  // [1]

    float* out      = (float*)d_out;
    float* outputs  = out;                                          // [T, N]
    float* states   = out + (size_t)T_STEPS * N_NEUR;               // [T, 2, N]
    float* decoded  = states + (size_t)T_STEPS * 2 * N_NEUR;        // [T, 1]

    float* p_part = (float*)d_ws;                  // [T, 32]  (1 MB)
    float* q      = p_part + (size_t)T_STEPS * 32; // [T]      (32 KB)

    // Phase A: GEMM (grid: N/128 x M/128)
    dim3 gA(N_NEUR / 128, T_STEPS / 128);
    snn_gemm_wmma<<<gA, 128, 0, stream>>>(X, W1, states);

    // Phase B: scan over time, 1024 neurons across 32 WGPs
    snn_scan<<<32, 32, 0, stream>>>(state, a_, b_, c_, d_, thresh, dt_, W2,
                                    outputs, states, p_part);

    // Phase C: decode reduction + serial scan
    snn_reduce_q<<<T_STEPS / 256, 256, 0, stream>>>(p_part, q);
    snn_dm_scan<<<1, 32, 0, stream>>>(q, leak, decoded);
}